// Mampa_23776938950764
// MI455X (gfx1250) — compile-verified
//
#include <hip/hip_runtime.h>
#include <hip/hip_bf16.h>
#include <math.h>

typedef __attribute__((ext_vector_type(2))) float v2f;
typedef __attribute__((ext_vector_type(8))) float v8f;

#define D_MODEL 1024
#define D_STATE 16
#define D_CONV  4
#define D_INNER 2048
#define DT_RANK 64
#define SEQLEN  2048

#define BM 128
#define BN 128
#define KB 32
#define KPAD (KB + 1)

// ---------------------------------------------------------------------------
// Generic fp32 WMMA GEMM:  C[M,N] = A[M,K] * B[N,K]^T   (all row-major)
// 256 threads = 8 waves. Block tile 128x128; wave tile 32x64
// (2x4 fragments of v_wmma_f32_16x16x4_f32). Register-staged, double-buffered
// LDS pipeline: one barrier per 32-deep K tile, global loads overlap compute.
// ---------------------------------------------------------------------------
__global__ __launch_bounds__(256) void gemm_abt_wmma(
    const float* __restrict__ A, int lda,
    const float* __restrict__ B, int ldb,
    float* __restrict__ C, int ldc,
    int M, int N, int K)
{
    __shared__ float As[2][BM][KPAD];
    __shared__ float Bs[2][BN][KPAD];

    const int tid  = threadIdx.x;
    const int bm   = blockIdx.y * BM;
    const int bn   = blockIdx.x * BN;
    const int wave = tid >> 5;            // 0..7
    const int lane = tid & 31;
    const int wm   = (wave >> 1) * 32;    // wave row offset (4 wave-rows)
    const int wn   = (wave & 1) * 64;     // wave col offset (2 wave-cols)
    const int ml   = lane & 15;           // M (resp. N) index inside fragment
    const int kh   = lane >> 4;           // K-half selector

    v8f acc[2][4] = {};                   // 32x64 of C per wave
    float ra[16], rb[16];                 // register staging (128x32 / 256 thr)

    const int nt = (K + KB - 1) / KB;

    // ---- prologue: stage tile 0 ----
    #pragma unroll
    for (int u = 0; u < 16; ++u) {
        int i = tid + u * 256;
        int r = i >> 5, c = i & 31;
        int ar = bm + r, br = bn + r;
        ra[u] = (ar < M && c < K) ? A[(size_t)ar * lda + c] : 0.0f;
        rb[u] = (br < N && c < K) ? B[(size_t)br * ldb + c] : 0.0f;
    }
    #pragma unroll
    for (int u = 0; u < 16; ++u) {
        int i = tid + u * 256;
        int r = i >> 5, c = i & 31;
        As[0][r][c] = ra[u];
        Bs[0][r][c] = rb[u];
    }
    __syncthreads();

    for (int kt = 0; kt < nt; ++kt) {
        const int cur = kt & 1;

        // ---- issue next tile's global loads (overlap with compute) ----
        if (kt + 1 < nt) {
            int k0 = (kt + 1) * KB;
            #pragma unroll
            for (int u = 0; u < 16; ++u) {
                int i = tid + u * 256;
                int r = i >> 5, c = i & 31;
                int gk = k0 + c;
                int ar = bm + r, br = bn + r;
                ra[u] = (ar < M && gk < K) ? A[(size_t)ar * lda + gk] : 0.0f;
                rb[u] = (br < N && gk < K) ? B[(size_t)br * ldb + gk] : 0.0f;
            }
        }
        // speculative prefetch of tile after next (global_prefetch)
        if (kt + 2 < nt) {
            int r = tid >> 1, sel = tid & 1;
            int arow = bm + r; if (arow >= M) arow = M - 1;
            int brow = bn + r; if (brow >= N) brow = N - 1;
            const float* p = sel ? &B[(size_t)brow * ldb + (kt + 2) * KB]
                                 : &A[(size_t)arow * lda + (kt + 2) * KB];
            __builtin_prefetch(p, 0, 0);
        }

        // ---- compute from LDS buffer `cur` ----
        {
            const float (*Asb)[KPAD] = As[cur];
            const float (*Bsb)[KPAD] = Bs[cur];
            #pragma unroll
            for (int kk = 0; kk < KB; kk += 4) {
                v2f a[2], b[4];
                #pragma unroll
                for (int i = 0; i < 2; ++i) {
                    // A 16x4: lanes 0-15 hold K+0/K+1, lanes 16-31 hold K+2/K+3
                    int row = wm + i * 16 + ml;
                    a[i][0] = Asb[row][kk + 2 * kh];
                    a[i][1] = Asb[row][kk + 2 * kh + 1];
                }
                #pragma unroll
                for (int j = 0; j < 4; ++j) {
                    int col = wn + j * 16 + ml;
                    b[j][0] = Bsb[col][kk + 2 * kh];
                    b[j][1] = Bsb[col][kk + 2 * kh + 1];
                }
                #pragma unroll
                for (int i = 0; i < 2; ++i)
                    #pragma unroll
                    for (int j = 0; j < 4; ++j)
                        acc[i][j] = __builtin_amdgcn_wmma_f32_16x16x4_f32(
                            false, a[i], false, b[j],
                            (short)0, acc[i][j], false, false);
            }
        }

        // ---- commit next tile to the other buffer; single barrier ----
        if (kt + 1 < nt) {
            #pragma unroll
            for (int u = 0; u < 16; ++u) {
                int i = tid + u * 256;
                int r = i >> 5, c = i & 31;
                As[cur ^ 1][r][c] = ra[u];
                Bs[cur ^ 1][r][c] = rb[u];
            }
            __syncthreads();
        }
    }

    // ---- store: C/D layout VGPR r -> M = r (lanes 0-15) / r+8 (16-31) ----
    #pragma unroll
    for (int i = 0; i < 2; ++i)
        #pragma unroll
        for (int j = 0; j < 4; ++j)
            #pragma unroll
            for (int r = 0; r < 8; ++r) {
                int m = bm + wm + i * 16 + 8 * kh + r;
                int n = bn + wn + j * 16 + ml;
                if (m < M && n < N)
                    C[(size_t)m * ldc + n] = acc[i][j][r];
            }
}

// ---------------------------------------------------------------------------
// Depthwise causal conv1d (kernel=4) + SiLU, reading x = xz[:, :D_INNER]
// ---------------------------------------------------------------------------
__global__ __launch_bounds__(256) void conv_silu_kernel(
    const float* __restrict__ xz,      // [L][2*D_INNER]
    const float* __restrict__ conv_w,  // [D_INNER][4]
    const float* __restrict__ conv_b,  // [D_INNER]
    float* __restrict__ x_act)         // [L][D_INNER]
{
    int idx = blockIdx.x * 256 + threadIdx.x;
    if (idx >= SEQLEN * D_INNER) return;
    int l = idx / D_INNER;
    int d = idx - l * D_INNER;
    float acc = conv_b[d];
    #pragma unroll
    for (int j = 0; j < D_CONV; ++j) {
        int ls = l - (D_CONV - 1) + j;
        if (ls >= 0)
            acc = fmaf(conv_w[d * D_CONV + j],
                       xz[(size_t)ls * (2 * D_INNER) + d], acc);
    }
    float s = 1.0f / (1.0f + __expf(-acc));
    x_act[idx] = acc * s;
}

// ---------------------------------------------------------------------------
// Selective scan, state-parallel: 16 lanes per channel (2 channels / wave32).
// Per step each lane: 1 exp + 2 fma for its state, then a width-16 shuffle
// reduction for y = sum_n h[n]*C[n].  Fused softplus bias, D-skip, silu(z).
// All per-step operands staged in LDS per 16-step chunk (coalesced reads).
// ---------------------------------------------------------------------------
__global__ __launch_bounds__(256) void scan_kernel(
    const float* __restrict__ xz,         // for z = xz[:, D_INNER:]
    const float* __restrict__ x_act,      // [L][D_INNER]
    const float* __restrict__ x_dbl,      // [L][96] (dt | B | C)
    const float* __restrict__ delta_raw,  // [L][D_INNER]
    const float* __restrict__ dt_proj_b,  // [D_INNER]
    const float* __restrict__ A_log,      // [D_INNER][16]
    const float* __restrict__ D_param,    // [D_INNER]
    float* __restrict__ ybuf)             // [L][D_INNER]
{
    __shared__ float bc[16][32];          // 16 steps x (B[16] | C[16])
    __shared__ float dstg[16][16];        // 16 steps x 16 channels: delta_raw
    __shared__ float xstg[16][16];        //                         x_act
    __shared__ float zstg[16][16];        //                         z

    const int tid = threadIdx.x;
    const int n   = tid & 15;             // state index (lane within group)
    const int dl  = tid >> 4;             // channel within block (0..15)
    const int d   = blockIdx.x * 16 + dl;

    float h   = 0.0f;
    float Ar  = -__expf(A_log[d * D_STATE + n]);
    float dtb = dt_proj_b[d];
    float Dp  = D_param[d];

    for (int l0 = 0; l0 < SEQLEN; l0 += 16) {
        __syncthreads();
        // stage B/C (shared across all channels)
        for (int i = tid; i < 16 * 32; i += 256) {
            int lo = i >> 5, c = i & 31;
            bc[lo][c] = x_dbl[(size_t)(l0 + lo) * 96 + DT_RANK + c];
        }
        // stage delta_raw / x_act / z for this block's 16 channels (coalesced)
        {
            int lo = tid >> 4, c = tid & 15;
            int dd = blockIdx.x * 16 + c;
            dstg[lo][c] = delta_raw[(size_t)(l0 + lo) * D_INNER + dd];
            xstg[lo][c] = x_act[(size_t)(l0 + lo) * D_INNER + dd];
            zstg[lo][c] = xz[(size_t)(l0 + lo) * (2 * D_INNER) + D_INNER + dd];
        }
        __syncthreads();

        #pragma unroll 4
        for (int lo = 0; lo < 16; ++lo) {
            float draw  = dstg[lo][dl] + dtb;
            float delta = (draw > 20.0f) ? draw : log1pf(__expf(draw));
            float xv = xstg[lo][dl];
            float dA = __expf(delta * Ar);
            h = fmaf(dA, h, delta * bc[lo][n] * xv);
            float p = h * bc[lo][16 + n];
            // reduce y over the 16 state lanes
            p += __shfl_xor(p, 1, 16);
            p += __shfl_xor(p, 2, 16);
            p += __shfl_xor(p, 4, 16);
            p += __shfl_xor(p, 8, 16);
            if (n == 0) {
                float zv = zstg[lo][dl];
                float zs = zv / (1.0f + __expf(-zv));    // silu(z)
                ybuf[(size_t)(l0 + lo) * D_INNER + d] = (p + Dp * xv) * zs;
            }
        }
    }
}

// ---------------------------------------------------------------------------
extern "C" void kernel_launch(void* const* d_in, const int* in_sizes, int n_in,
                              void* d_out, int out_size, void* d_ws, size_t ws_size,
                              hipStream_t stream) {
    (void)in_sizes; (void)n_in; (void)out_size; (void)ws_size;

    const float* hidden   = (const float*)d_in[0];
    const float* in_w     = (const float*)d_in[1];  // [4096][1024]
    const float* conv_w   = (const float*)d_in[2];  // [2048][1][4]
    const float* conv_b   = (const float*)d_in[3];
    const float* xproj_w  = (const float*)d_in[4];  // [96][2048]
    const float* dtproj_w = (const float*)d_in[5];  // [2048][64]
    const float* dtproj_b = (const float*)d_in[6];
    const float* out_w    = (const float*)d_in[7];  // [1024][2048]
    const float* A_log    = (const float*)d_in[8];
    const float* D_param  = (const float*)d_in[9];
    float* out = (float*)d_out;

    // workspace layout (floats)
    float* ws        = (float*)d_ws;
    float* xz        = ws;                                        // 2048*4096
    float* x_act     = xz        + (size_t)SEQLEN * 2 * D_INNER;  // 2048*2048
    float* x_dbl     = x_act     + (size_t)SEQLEN * D_INNER;      // 2048*96
    float* delta_raw = x_dbl     + (size_t)SEQLEN * 96;           // 2048*2048
    float* ybuf      = delta_raw + (size_t)SEQLEN * D_INNER;      // 2048*2048

    dim3 blk(256);

    // 1) xz = H * Win^T         (M=2048, N=4096, K=1024)
    gemm_abt_wmma<<<dim3((2 * D_INNER + BN - 1) / BN, (SEQLEN + BM - 1) / BM),
                    blk, 0, stream>>>(
        hidden, D_MODEL, in_w, D_MODEL, xz, 2 * D_INNER,
        SEQLEN, 2 * D_INNER, D_MODEL);

    // 2) x_act = silu(causal_conv(x))
    conv_silu_kernel<<<(SEQLEN * D_INNER) / 256, 256, 0, stream>>>(
        xz, conv_w, conv_b, x_act);

    // 3) x_dbl = x_act * x_proj_w^T   (M=2048, N=96, K=2048)
    gemm_abt_wmma<<<dim3((96 + BN - 1) / BN, (SEQLEN + BM - 1) / BM),
                    blk, 0, stream>>>(
        x_act, D_INNER, xproj_w, D_INNER, x_dbl, 96,
        SEQLEN, 96, D_INNER);

    // 4) delta_raw = dt * dt_proj_w^T (A = x_dbl[:, :64] via lda=96; K=64)
    gemm_abt_wmma<<<dim3((D_INNER + BN - 1) / BN, (SEQLEN + BM - 1) / BM),
                    blk, 0, stream>>>(
        x_dbl, 96, dtproj_w, DT_RANK, delta_raw, D_INNER,
        SEQLEN, D_INNER, DT_RANK);

    // 5) fused selective scan (state-parallel) + gating
    scan_kernel<<<D_INNER / 16, 256, 0, stream>>>(
        xz, x_act, x_dbl, delta_raw, dtproj_b, A_log, D_param, ybuf);

    // 6) out = y * out_proj_w^T  (M=2048, N=1024, K=2048)
    gemm_abt_wmma<<<dim3((D_MODEL + BN - 1) / BN, (SEQLEN + BM - 1) / BM),
                    blk, 0, stream>>>(
        ybuf, D_INNER, out_w, D_INNER, out, D_MODEL,
        SEQLEN, D_MODEL, D_INNER);
}